// MultiHeadAttention_1580547966848
// MI455X (gfx1250) — compile-verified
//
#include <hip/hip_runtime.h>
#include <hip/hip_bf16.h>

typedef __attribute__((ext_vector_type(2))) float v2f;
typedef __attribute__((ext_vector_type(8))) float v8f;

#define BATCH 2
#define SEQ   2048
#define DMODEL 1024
#define NHEADS 16
#define DK    64
#define HDK   1024
#define NEGINF (-1.0e9f)

static __device__ __forceinline__ v8f wmma_f32(v2f a, v2f b, v8f c) {
    return __builtin_amdgcn_wmma_f32_16x16x4_f32(false, a, false, b, (short)0, c,
                                                 false, false);
}

// Generic GEMM: out = A(MxK) @ W(KxN) + bias, fp32 WMMA.
// layout 0: scatter to head-major (B,H,S,DK); layout 1: row-major (M,N).
// Block = 128 threads (4 waves); wave w -> rows m0..m0+15, block covers 64 rows x 64 cols.
__global__ void proj_gemm(const float* __restrict__ A, const float* __restrict__ W,
                          const float* __restrict__ bias, float* __restrict__ out,
                          int K, int N, int layout) {
    const int lane = threadIdx.x & 31;
    const int wave = threadIdx.x >> 5;
    const int hl   = lane >> 4;     // half of wave
    const int idx  = lane & 15;
    const int m0 = blockIdx.x * 64 + wave * 16;
    const int n0 = blockIdx.y * 64;

    v8f acc[4] = {};
    const float* arow = A + (size_t)(m0 + idx) * K;

    for (int k = 0; k < K; k += 4) {
        v2f a;
        a.x = arow[k + 2 * hl];
        a.y = arow[k + 2 * hl + 1];
#pragma unroll
        for (int j = 0; j < 4; ++j) {
            const int n = n0 + j * 16 + idx;
            v2f b;
            b.x = W[(size_t)(k + 2 * hl) * N + n];
            b.y = W[(size_t)(k + 2 * hl + 1) * N + n];
            acc[j] = wmma_f32(a, b, acc[j]);
        }
    }

#pragma unroll
    for (int j = 0; j < 4; ++j) {
        const int n = n0 + j * 16 + idx;
        const float bv = bias[n];
#pragma unroll
        for (int v = 0; v < 8; ++v) {
            const int m = m0 + v + 8 * hl;
            const float val = acc[j][v] + bv;
            if (layout == 0) {
                const int b = m / SEQ, s = m % SEQ;
                const int h = n >> 6, d = n & 63;
                out[((((size_t)b * NHEADS + h) * SEQ + s) * DK) + d] = val;
            } else {
                out[(size_t)m * N + n] = val;
            }
        }
    }
}

// scores = (qh @ kh^T) / sqrt(dk), masked. Wave -> 16 q-rows x 64 kv-cols.
// grid: (Sq/16, Skv/256, B*H), block 128 (4 waves along kv).
__global__ void scores_kernel(const float* __restrict__ qh, const float* __restrict__ kh,
                              const int* __restrict__ mask, float* __restrict__ scores) {
    const int lane = threadIdx.x & 31;
    const int wave = threadIdx.x >> 5;
    const int hl = lane >> 4;
    const int idx = lane & 15;
    const int bh = blockIdx.z;
    const int b = bh >> 4;
    const int q0 = blockIdx.x * 16;
    const int n0 = blockIdx.y * 256 + wave * 64;

    const float* Q  = qh + (size_t)bh * SEQ * DK;
    const float* Km = kh + (size_t)bh * SEQ * DK;
    const float* qrow = Q + (size_t)(q0 + idx) * DK;

    v8f acc[4] = {};
#pragma unroll 4
    for (int k = 0; k < DK; k += 4) {
        v2f a;
        a.x = qrow[k + 2 * hl];
        a.y = qrow[k + 2 * hl + 1];
#pragma unroll
        for (int j = 0; j < 4; ++j) {
            const float* krow = Km + (size_t)(n0 + j * 16 + idx) * DK + k + 2 * hl;
            v2f bb;
            bb.x = krow[0];
            bb.y = krow[1];
            acc[j] = wmma_f32(a, bb, acc[j]);
        }
    }

    const float scale = 0.125f;  // 1/sqrt(64)
#pragma unroll
    for (int j = 0; j < 4; ++j) {
        const int n = n0 + j * 16 + idx;
        const bool keep = mask[b * SEQ + n] != 0;
        float* srow = scores + ((size_t)bh * SEQ + q0) * SEQ + n;
#pragma unroll
        for (int v = 0; v < 8; ++v) {
            const float val = keep ? acc[j][v] * scale : NEGINF;
            srow[(size_t)(v + 8 * hl) * SEQ] = val;
        }
    }
}

// Per-row max and sum(exp(s-max)). One wave per row; block 256 = 8 rows.
__global__ void softmax_stats(const float* __restrict__ scores,
                              float* __restrict__ mrow, float* __restrict__ lrow) {
    const int lane = threadIdx.x & 31;
    const int wave = threadIdx.x >> 5;
    const size_t row = (size_t)blockIdx.x * 8 + wave;
    const float* s = scores + row * SEQ;

    float m = -3.0e38f;
    for (int i = lane; i < SEQ; i += 32) m = fmaxf(m, s[i]);
#pragma unroll
    for (int off = 16; off > 0; off >>= 1) m = fmaxf(m, __shfl_xor(m, off, 32));

    float l = 0.0f;
    for (int i = lane; i < SEQ; i += 32) l += __expf(s[i] - m);
#pragma unroll
    for (int off = 16; off > 0; off >>= 1) l += __shfl_xor(l, off, 32);

    if (lane == 0) { mrow[row] = m; lrow[row] = l; }
}

// attn = softmax(scores) @ vh, written directly into concat (B,Sq,H*DK).
// One wave per block; wave -> 16 q-rows x 64 (full dk). grid (Sq/16, 1, B*H).
__global__ void attn_pv(const float* __restrict__ scores, const float* __restrict__ vh,
                        const float* __restrict__ mrow, const float* __restrict__ lrow,
                        float* __restrict__ concat) {
    const int lane = threadIdx.x;
    const int hl = lane >> 4;
    const int idx = lane & 15;
    const int bh = blockIdx.z;
    const int b = bh >> 4, h = bh & 15;
    const int q0 = blockIdx.x * 16;

    const float* S = scores + ((size_t)bh * SEQ + q0) * SEQ;
    const float* V = vh + (size_t)bh * SEQ * DK;
    const float mr = mrow[(size_t)bh * SEQ + q0 + idx];  // A-operand row = idx
    const float* srow = S + (size_t)idx * SEQ;

    v8f acc[4] = {};
    for (int k = 0; k < SEQ; k += 4) {
        v2f a;
        a.x = __expf(srow[k + 2 * hl] - mr);
        a.y = __expf(srow[k + 2 * hl + 1] - mr);
#pragma unroll
        for (int j = 0; j < 4; ++j) {
            const float* vr = V + (size_t)(k + 2 * hl) * DK + j * 16 + idx;
            v2f bb;
            bb.x = vr[0];
            bb.y = vr[DK];
            acc[j] = wmma_f32(a, bb, acc[j]);
        }
    }

#pragma unroll
    for (int v = 0; v < 8; ++v) {
        const int q = q0 + v + 8 * hl;
        const float inv = 1.0f / lrow[(size_t)bh * SEQ + q];
        float* orow = concat + ((size_t)b * SEQ + q) * HDK + h * DK;
#pragma unroll
        for (int j = 0; j < 4; ++j) orow[j * 16 + idx] = acc[v >= 0 ? j : j][v] * inv;
    }
}

extern "C" void kernel_launch(void* const* d_in, const int* in_sizes, int n_in,
                              void* d_out, int out_size, void* d_ws, size_t ws_size,
                              hipStream_t stream) {
    (void)in_sizes; (void)n_in; (void)out_size; (void)ws_size;
    const float* q  = (const float*)d_in[0];
    const float* k  = (const float*)d_in[1];
    const float* v  = (const float*)d_in[2];
    const int*  msk = (const int*)d_in[3];
    const float* Wq = (const float*)d_in[4];
    const float* bq = (const float*)d_in[5];
    const float* Wk = (const float*)d_in[6];
    const float* bk = (const float*)d_in[7];
    const float* Wv = (const float*)d_in[8];
    const float* bv = (const float*)d_in[9];
    const float* Wo = (const float*)d_in[10];
    const float* bo = (const float*)d_in[11];

    float* out    = (float*)d_out;                       // (B,Sq,D)   4,194,304 f
    float* scores = out + (size_t)BATCH * SEQ * DMODEL;  // (B,H,Sq,Skv) 134,217,728 f

    float* ws = (float*)d_ws;
    const size_t nBH = (size_t)BATCH * NHEADS * SEQ * DK;  // 4,194,304
    float* qh   = ws;
    float* kh   = qh + nBH;
    float* vhh  = kh + nBH;
    float* conc = vhh + nBH;
    float* mrow = conc + (size_t)BATCH * SEQ * HDK;
    float* lrow = mrow + (size_t)BATCH * NHEADS * SEQ;

    const dim3 gp(BATCH * SEQ / 64, DMODEL / 64);  // 64 x 16 blocks, 128 thr
    proj_gemm<<<gp, 128, 0, stream>>>(q, Wq, bq, qh, DMODEL, HDK, 0);
    proj_gemm<<<gp, 128, 0, stream>>>(k, Wk, bk, kh, DMODEL, HDK, 0);
    proj_gemm<<<gp, 128, 0, stream>>>(v, Wv, bv, vhh, DMODEL, HDK, 0);

    scores_kernel<<<dim3(SEQ / 16, SEQ / 256, BATCH * NHEADS), 128, 0, stream>>>(
        qh, kh, msk, scores);

    softmax_stats<<<dim3((BATCH * NHEADS * SEQ) / 8), 256, 0, stream>>>(
        scores, mrow, lrow);

    attn_pv<<<dim3(SEQ / 16, 1, BATCH * NHEADS), 32, 0, stream>>>(
        scores, vhh, mrow, lrow, conc);

    proj_gemm<<<gp, 128, 0, stream>>>(conc, Wo, bo, out, HDK, DMODEL, 1);
}